// TemporalGraphNetwork_51917564674486
// MI455X (gfx1250) — compile-verified
//
#include <hip/hip_runtime.h>
#include <hip/hip_bf16.h>
#include <math.h>

#define NNODE 50000
#define NEDGE 500000
#define NNEV 1000
#define NEEV 1000
#define MEMD 64
#define TENCD 32
#define EVEMB 64
#define MSGD 232           // 8 + 64 + 64 + 32 + 64
#define HOD 128
#define OUTD 64
#define NEVT_TOTAL (NNEV + 2 * NEEV)

typedef __attribute__((ext_vector_type(2))) float v2f;
typedef __attribute__((ext_vector_type(8))) float v8f;
typedef __attribute__((ext_vector_type(4))) int   v4i;

// ---------- optional gfx1250 async global->LDS copy path ----------
#if defined(__has_builtin)
#if __has_builtin(__builtin_amdgcn_global_load_async_to_lds_b128)
#define HAVE_ASYNC_COPY 1
#endif
#endif

#ifdef HAVE_ASYNC_COPY
typedef __attribute__((address_space(1))) v4i* g_v4i_p;   // global
typedef __attribute__((address_space(3))) v4i* l_v4i_p;   // LDS
static __device__ __forceinline__ void async_cp16(const float* g, float* l) {
  v4i* gg = (v4i*)const_cast<float*>(g);  // pointee reinterpret (generic AS)
  v4i* ll = (v4i*)l;
  __builtin_amdgcn_global_load_async_to_lds_b128((g_v4i_p)gg, (l_v4i_p)ll, 0, 0);
}
static __device__ __forceinline__ void async_wait() {
  asm volatile("s_wait_asynccnt 0x0" ::: "memory");
}
#endif

// ---------- ordered-uint encoding for float atomicMax ----------
static __device__ __forceinline__ unsigned f2ord(float f) {
  unsigned b = __float_as_uint(f);
  return (b & 0x80000000u) ? ~b : (b | 0x80000000u);
}
static __device__ __forceinline__ float ord2f(unsigned k) {
  unsigned b = (k & 0x80000000u) ? (k & 0x7fffffffu) : ~k;
  return __uint_as_float(b);
}

// ---------- fp32 WMMA GEMM: C[M,N] = A[M,K] @ B[K,N] (+bias) ----------
// N is compile-time (64 or 128) so staging index math is shifts, not idiv.
// grid.x = M/16, block = 256 (8 waves); wave w owns N-tile w.
template <int N>
__global__ void wmma_gemm_bias(const float* __restrict__ A, const float* __restrict__ B,
                               const float* __restrict__ bias, float* __restrict__ C,
                               int K) {
  __shared__ float lA[16 * 32];
  __shared__ float lB[32 * 128];
  const int tid = threadIdx.x;
  const int wave = tid >> 5;
  const int lane = tid & 31;
  const int m0 = blockIdx.x * 16;
  const int n0 = wave * 16;
  v8f acc = {};
  for (int kc = 0; kc < K; kc += 32) {
    __syncthreads();
#ifdef HAVE_ASYNC_COPY
    // A chunk: 16x32 floats = 128 x b128; threads 0..127 issue one each.
    if (tid < 128) {
      int r = tid >> 3, c4 = (tid & 7) << 2;
      async_cp16(&A[(size_t)(m0 + r) * K + kc + c4], &lA[r * 32 + c4]);
    }
    // B chunk: 32xN floats = 8N x b128.
    for (int t = tid; t < 8 * N; t += 256) {
      int r = t / (N / 4);                 // compile-time N -> shift
      int c4 = (t - r * (N / 4)) << 2;
      async_cp16(&B[(size_t)(kc + r) * N + c4], &lB[r * 128 + c4]);
    }
    async_wait();
#else
    for (int idx = tid; idx < 16 * 32; idx += 256) {
      int r = idx >> 5, c = idx & 31;
      lA[idx] = A[(size_t)(m0 + r) * K + kc + c];
      if (kc + 32 < K) __builtin_prefetch(&A[(size_t)(m0 + r) * K + kc + 32 + c], 0, 0);
    }
    for (int idx = tid; idx < 32 * N; idx += 256) {
      int r = idx / N;                     // compile-time N -> shift
      int c = idx - r * N;
      lB[r * 128 + c] = B[(size_t)(kc + r) * N + c];
    }
#endif
    __syncthreads();
    if (n0 < N) {
      const int m = lane & 15;
      const int kh = (lane >> 4) * 2;      // lane covers K = kh, kh+1 within each group of 4
      const int nn = n0 + (lane & 15);
      for (int k4 = 0; k4 < 32; k4 += 4) {
        v2f a, b;
        a.x = lA[m * 32 + k4 + kh];
        a.y = lA[m * 32 + k4 + kh + 1];
        b.x = lB[(k4 + kh) * 128 + nn];
        b.y = lB[(k4 + kh + 1) * 128 + nn];
        acc = __builtin_amdgcn_wmma_f32_16x16x4_f32(false, a, false, b, (short)0, acc,
                                                    false, false);
      }
    }
  }
  if (n0 < N) {
    const int n = lane & 15;
    const int rb = (lane >> 4) * 8;        // C layout: vgpr g -> row g (lanes<16) / g+8
    const float bv = bias ? bias[n0 + n] : 0.0f;
    for (int g = 0; g < 8; ++g)
      C[(size_t)(m0 + g + rb) * N + n0 + n] = acc[g] + bv;
  }
}

// ---------- edge-attr GEMM with fused time encoder ----------
// EE[E,128] = concat(cos(rel_t * tw + tb), edge_feat)[E,96] @ W[96,128]
__global__ void ee_gemm(const float* __restrict__ ets, const float* __restrict__ elu,
                        const float* __restrict__ efeat, const float* __restrict__ tw,
                        const float* __restrict__ tb, const float* __restrict__ W,
                        float* __restrict__ EE, int E) {
  __shared__ float lA[16 * 96];
  __shared__ float lB[96 * 128];
  const int tid = threadIdx.x;
  const int wave = tid >> 5;
  const int lane = tid & 31;
  const int e0 = blockIdx.x * 16;
#ifdef HAVE_ASYNC_COPY
  // W: 96*128 floats = 3072 x b128; 12 per thread.
  for (int t = tid; t < 3072; t += 256) async_cp16(&W[t << 2], &lB[t << 2]);
#else
  for (int idx = tid; idx < 96 * 128; idx += 256) lB[idx] = W[idx];
#endif
  for (int idx = tid; idx < 16 * 96; idx += 256) {
    int r = idx / 96, c = idx - r * 96;
    int e = e0 + r;
    float v;
    if (c < TENCD) {
      float rel = ets[e] - elu[e];
      v = cosf(rel * tw[c] + tb[c]);
    } else {
      v = efeat[(size_t)e * EVEMB + (c - TENCD)];
    }
    lA[idx] = v;
  }
#ifdef HAVE_ASYNC_COPY
  async_wait();
#endif
  __syncthreads();
  const int m = lane & 15;
  const int kh = (lane >> 4) * 2;
  const int nn = wave * 16 + (lane & 15);
  v8f acc = {};
  for (int k4 = 0; k4 < 96; k4 += 4) {
    v2f a, b;
    a.x = lA[m * 96 + k4 + kh];
    a.y = lA[m * 96 + k4 + kh + 1];
    b.x = lB[(k4 + kh) * 128 + nn];
    b.y = lB[(k4 + kh + 1) * 128 + nn];
    acc = __builtin_amdgcn_wmma_f32_16x16x4_f32(false, a, false, b, (short)0, acc,
                                                false, false);
  }
  const int rb = (lane >> 4) * 8;
  for (int g = 0; g < 8; ++g)
    EE[(size_t)(e0 + g + rb) * 128 + nn] = acc[g];
}

// ---------- event message scatter + touched compaction ----------
__global__ void msg_scatter(const int* __restrict__ ne_type, const int* __restrict__ ne_node,
                            const float* __restrict__ ne_emb, const float* __restrict__ ne_ts,
                            const float* __restrict__ ne_mask, const int* __restrict__ ev_type,
                            const int* __restrict__ ev_src, const int* __restrict__ ev_dst,
                            const int* __restrict__ ev_eid, const float* __restrict__ ev_emb,
                            const float* __restrict__ ev_ts, const float* __restrict__ ev_mask,
                            const float* __restrict__ memory, const float* __restrict__ elu,
                            const float* __restrict__ tyemb, const float* __restrict__ tw,
                            const float* __restrict__ tb, float* __restrict__ agg,
                            int* __restrict__ flags, int* __restrict__ tlist) {
  int ev = blockIdx.x * 8 + (threadIdx.x >> 5);
  if (ev >= NEVT_TOTAL) return;
  const int lane = threadIdx.x & 31;
  int node, type;
  float mask, tval;
  const float *ma, *mb, *emb;
  if (ev < NNEV) {
    node = ne_node[ev]; type = ne_type[ev]; mask = ne_mask[ev]; tval = ne_ts[ev];
    ma = memory + (size_t)node * MEMD; mb = nullptr;
    emb = ne_emb + (size_t)ev * EVEMB;
  } else {
    int e = ev - NNEV;
    bool first = e < NEEV;
    if (!first) e -= NEEV;
    int s = ev_src[e], d = ev_dst[e];
    node = first ? s : d;
    type = ev_type[e]; mask = ev_mask[e];
    tval = ev_ts[e] - elu[ev_eid[e]];
    ma = memory + (size_t)(first ? s : d) * MEMD;
    mb = memory + (size_t)(first ? d : s) * MEMD;
    emb = ev_emb + (size_t)e * EVEMB;
  }
  for (int c = lane; c < MSGD; c += 32) {
    float val;
    if (c < 8)        val = tyemb[type * 8 + c];
    else if (c < 72)  val = ma[c - 8];
    else if (c < 136) val = mb ? mb[c - 72] : 0.0f;
    else if (c < 168) val = cosf(tval * tw[c - 136] + tb[c - 136]);
    else              val = emb[c - 168];
    atomicAdd(&agg[(size_t)node * MSGD + c], val * mask);
  }
  if (lane == 0) {
    if (atomicCAS(&flags[node], 0, 1) == 0) {
      int i = atomicAdd(&tlist[0], 1);
      tlist[1 + i] = node;
    }
  }
}

// ---------- x0 = [node_features | memory] ----------
__global__ void build_x0(const float* __restrict__ nf, const float* __restrict__ mem,
                         float* __restrict__ x0) {
  size_t idx = (size_t)blockIdx.x * 256 + threadIdx.x;
  if (idx >= (size_t)NNODE * 128) return;
  int n = (int)(idx >> 7), c = (int)(idx & 127);
  x0[idx] = (c < 64) ? nf[(size_t)n * 64 + c] : mem[(size_t)n * 64 + (c - 64)];
}

// ---------- GRU on touched nodes, writes memory half of x0 ----------
__global__ void gru_touched(const float* __restrict__ agg, const float* __restrict__ memory,
                            const float* __restrict__ w_ih, const float* __restrict__ w_hh,
                            const float* __restrict__ b_ih, const float* __restrict__ b_hh,
                            const int* __restrict__ tlist, float* __restrict__ x0) {
  if (blockIdx.x >= tlist[0]) return;
  const int node = tlist[1 + blockIdx.x];
  __shared__ float sx[MSGD];
  __shared__ float sh[MEMD];
  __shared__ float gi[192];
  __shared__ float gh[192];
  const int tid = threadIdx.x;
  for (int c = tid; c < MSGD; c += 256) sx[c] = agg[(size_t)node * MSGD + c];
  for (int c = tid; c < MEMD; c += 256) sh[c] = memory[(size_t)node * MEMD + c];
  __syncthreads();
  if (tid < 192) {
    float a = b_ih[tid];
    const float* w = w_ih + (size_t)tid * MSGD;
    for (int k = 0; k < MSGD; ++k) a += w[k] * sx[k];
    gi[tid] = a;
    float g = b_hh[tid];
    const float* wh = w_hh + (size_t)tid * MEMD;
    for (int k = 0; k < MEMD; ++k) g += wh[k] * sh[k];
    gh[tid] = g;
  }
  __syncthreads();
  if (tid < 64) {
    float r = 1.0f / (1.0f + expf(-(gi[tid] + gh[tid])));
    float z = 1.0f / (1.0f + expf(-(gi[64 + tid] + gh[64 + tid])));
    float nn = tanhf(gi[128 + tid] + r * gh[128 + tid]);
    x0[(size_t)node * 128 + 64 + tid] = (1.0f - z) * nn + z * sh[tid];
  }
}

// ---------- xin = [h_prev | x0] ----------
__global__ void concat_xin(const float* __restrict__ h, const float* __restrict__ x0,
                           float* __restrict__ xin) {
  size_t idx = (size_t)blockIdx.x * 256 + threadIdx.x;
  if (idx >= (size_t)NNODE * 256) return;
  int n = (int)(idx >> 8), c = (int)(idx & 255);
  xin[idx] = (c < 128) ? h[(size_t)n * 128 + c] : x0[(size_t)n * 128 + (c - 128)];
}

// ---------- attention pass A: alpha + running max (one wave per edge) ----------
__global__ void edge_alpha(const int* __restrict__ src, const int* __restrict__ dst,
                           const float* __restrict__ Q, const float* __restrict__ Kb,
                           const float* __restrict__ EE, float* __restrict__ alpha,
                           unsigned* __restrict__ amax, int E) {
  int e = blockIdx.x * 8 + (threadIdx.x >> 5);
  if (e >= E) return;
  const int lane = threadIdx.x & 31;
  const int s = src[e], d = dst[e];
  float4 q = ((const float4*)(Q + (size_t)d * 128))[lane];
  float4 k = ((const float4*)(Kb + (size_t)s * 128))[lane];
  float4 ee = ((const float4*)(EE + (size_t)e * 128))[lane];
  float p = q.x * (k.x + ee.x) + q.y * (k.y + ee.y) + q.z * (k.z + ee.z) + q.w * (k.w + ee.w);
  p += __shfl_xor(p, 1, 32);
  p += __shfl_xor(p, 2, 32);
  p += __shfl_xor(p, 4, 32);
  p += __shfl_xor(p, 8, 32);               // reduced within each 16-lane half (one per head)
  float a = p * 0.125f;                    // 1/sqrt(64)
  if ((lane & 15) == 0) {
    int h = lane >> 4;
    alpha[(size_t)e * 2 + h] = a;
    atomicMax(&amax[(size_t)d * 2 + h], f2ord(a));
  }
}

// ---------- attention pass B: exp, denom, numerator scatter ----------
__global__ void edge_accum(const int* __restrict__ src, const int* __restrict__ dst,
                           const float* __restrict__ V, const float* __restrict__ EE,
                           const float* __restrict__ alpha, const unsigned* __restrict__ amax,
                           float* __restrict__ asum, float* __restrict__ hsum, int E) {
  int e = blockIdx.x * 8 + (threadIdx.x >> 5);
  if (e >= E) return;
  const int lane = threadIdx.x & 31;
  const int s = src[e], d = dst[e];
  const int h = lane >> 4;
  float mx = ord2f(amax[(size_t)d * 2 + h]);
  float ex = expf(alpha[(size_t)e * 2 + h] - mx);
  if ((lane & 15) == 0) atomicAdd(&asum[(size_t)d * 2 + h], ex);
  float4 v = ((const float4*)(V + (size_t)s * 128))[lane];
  float4 ee = ((const float4*)(EE + (size_t)e * 128))[lane];
  float* o = hsum + (size_t)d * 128 + 4 * lane;
  atomicAdd(o + 0, (v.x + ee.x) * ex);
  atomicAdd(o + 1, (v.y + ee.y) * ex);
  atomicAdd(o + 2, (v.z + ee.z) * ex);
  atomicAdd(o + 3, (v.w + ee.w) * ex);
}

// ---------- h = hsum / (asum + eps) + skip ----------
__global__ void combine_out(const float* __restrict__ hsum, const float* __restrict__ asum,
                            const float* __restrict__ skip, float* __restrict__ H) {
  size_t idx = (size_t)blockIdx.x * 256 + threadIdx.x;
  if (idx >= (size_t)NNODE * 128) return;
  int n = (int)(idx >> 7), c = (int)(idx & 127);
  H[idx] = hsum[idx] / (asum[(size_t)n * 2 + (c >> 6)] + 1e-16f) + skip[idx];
}

extern "C" void kernel_launch(void* const* d_in, const int* in_sizes, int n_in,
                              void* d_out, int out_size, void* d_ws, size_t ws_size,
                              hipStream_t stream) {
  (void)n_in; (void)out_size; (void)ws_size;
  const int*   ne_type = (const int*)d_in[0];
  const int*   ne_node = (const int*)d_in[1];
  const float* ne_emb  = (const float*)d_in[2];
  const float* ne_ts   = (const float*)d_in[3];
  const float* ne_mask = (const float*)d_in[4];
  const int*   ev_type = (const int*)d_in[5];
  const int*   ev_src  = (const int*)d_in[6];
  const int*   ev_dst  = (const int*)d_in[7];
  const int*   ev_eid  = (const int*)d_in[8];
  const float* ev_emb  = (const float*)d_in[9];
  const float* ev_ts   = (const float*)d_in[10];
  const float* ev_mask = (const float*)d_in[11];
  const float* memory  = (const float*)d_in[12];
  const float* nfeat   = (const float*)d_in[13];
  const int*   eindex  = (const int*)d_in[14];
  const float* efeat   = (const float*)d_in[15];
  const float* ets     = (const float*)d_in[16];
  const float* elu     = (const float*)d_in[17];
  const float* tyemb   = (const float*)d_in[18];
  const float* tw      = (const float*)d_in[19];
  const float* tb      = (const float*)d_in[20];
  const float* w_ih    = (const float*)d_in[21];
  const float* w_hh    = (const float*)d_in[22];
  const float* b_ih    = (const float*)d_in[23];
  const float* b_hh    = (const float*)d_in[24];
  const int gbase = 25;
  const float* lin_w = (const float*)d_in[gbase + 27];
  const float* lin_b = (const float*)d_in[gbase + 28];

  // gnn_params leaf order auto-detect: alphabetical pytree (bk first, size 128)
  // vs dict-insertion (wq first, size 16384).
  const bool sorted = in_sizes[gbase] <= 1024;
  const float *WQ[3], *BQ[3], *WK[3], *BK[3], *WV[3], *BV[3], *WE[3], *WS[3], *BS[3];
  for (int b = 0; b < 3; ++b) {
    void* const* p = d_in + gbase + b * 9;
    if (sorted) {  // bk,bq,bs,bv,we,wk,wq,ws,wv
      BK[b]=(const float*)p[0]; BQ[b]=(const float*)p[1]; BS[b]=(const float*)p[2];
      BV[b]=(const float*)p[3]; WE[b]=(const float*)p[4]; WK[b]=(const float*)p[5];
      WQ[b]=(const float*)p[6]; WS[b]=(const float*)p[7]; WV[b]=(const float*)p[8];
    } else {       // wq,bq,wk,bk,wv,bv,we,ws,bs
      WQ[b]=(const float*)p[0]; BQ[b]=(const float*)p[1]; WK[b]=(const float*)p[2];
      BK[b]=(const float*)p[3]; WV[b]=(const float*)p[4]; BV[b]=(const float*)p[5];
      WE[b]=(const float*)p[6]; WS[b]=(const float*)p[7]; BS[b]=(const float*)p[8];
    }
  }

  // workspace carve
  char* base = (char*)d_ws;
  size_t off = 0;
  auto carve = [&](size_t bytes) -> void* {
    void* p = base + off;
    off += (bytes + 255) & ~(size_t)255;
    return p;
  };
  float*    agg   = (float*)carve((size_t)NNODE * MSGD * 4);
  float*    x0    = (float*)carve((size_t)NNODE * 128 * 4);
  float*    xin   = (float*)carve((size_t)NNODE * 256 * 4);
  float*    qb    = (float*)carve((size_t)NNODE * 128 * 4);
  float*    kb    = (float*)carve((size_t)NNODE * 128 * 4);
  float*    vb    = (float*)carve((size_t)NNODE * 128 * 4);
  float*    skip  = (float*)carve((size_t)NNODE * 128 * 4);
  float*    hprev = (float*)carve((size_t)NNODE * 128 * 4);
  float*    hsum  = (float*)carve((size_t)NNODE * 128 * 4);
  float*    eebuf = (float*)carve((size_t)NEDGE * 128 * 4);
  float*    alpha = (float*)carve((size_t)NEDGE * 2 * 4);
  unsigned* amax  = (unsigned*)carve((size_t)NNODE * 2 * 4);
  float*    asum  = (float*)carve((size_t)NNODE * 2 * 4);
  int*      flags = (int*)carve((size_t)NNODE * 4);
  int*      tlist = (int*)carve((size_t)(NEVT_TOTAL + 1) * 4);

  const int* gsrc = eindex;
  const int* gdst = eindex + NEDGE;

  // ---- phase 1: messages + GRU memory update ----
  (void)hipMemsetAsync(agg, 0, (size_t)NNODE * MSGD * 4, stream);
  (void)hipMemsetAsync(flags, 0, (size_t)NNODE * 4, stream);
  (void)hipMemsetAsync(tlist, 0, (size_t)(NEVT_TOTAL + 1) * 4, stream);
  msg_scatter<<<(NEVT_TOTAL + 7) / 8, 256, 0, stream>>>(
      ne_type, ne_node, ne_emb, ne_ts, ne_mask, ev_type, ev_src, ev_dst, ev_eid,
      ev_emb, ev_ts, ev_mask, memory, elu, tyemb, tw, tb, agg, flags, tlist);
  build_x0<<<(NNODE * 128 + 255) / 256, 256, 0, stream>>>(nfeat, memory, x0);
  gru_touched<<<NEVT_TOTAL, 256, 0, stream>>>(agg, memory, w_ih, w_hh, b_ih, b_hh,
                                              tlist, x0);

  // ---- phase 2: 3 TransformerConv blocks ----
  for (int i = 0; i < 3; ++i) {
    const float* A = (i == 0) ? x0 : xin;
    const int Kdim = (i == 0) ? 128 : 256;
    if (i > 0)
      concat_xin<<<(NNODE * 256 + 255) / 256, 256, 0, stream>>>(hprev, x0, xin);
    wmma_gemm_bias<128><<<NNODE / 16, 256, 0, stream>>>(A, WQ[i], BQ[i], qb, Kdim);
    wmma_gemm_bias<128><<<NNODE / 16, 256, 0, stream>>>(A, WK[i], BK[i], kb, Kdim);
    wmma_gemm_bias<128><<<NNODE / 16, 256, 0, stream>>>(A, WV[i], BV[i], vb, Kdim);
    wmma_gemm_bias<128><<<NNODE / 16, 256, 0, stream>>>(A, WS[i], BS[i], skip, Kdim);
    ee_gemm<<<NEDGE / 16, 256, 0, stream>>>(ets, elu, efeat, tw, tb, WE[i], eebuf, NEDGE);
    (void)hipMemsetAsync(amax, 0, (size_t)NNODE * 2 * 4, stream);  // ord-encoding bottom
    (void)hipMemsetAsync(asum, 0, (size_t)NNODE * 2 * 4, stream);
    (void)hipMemsetAsync(hsum, 0, (size_t)NNODE * 128 * 4, stream);
    edge_alpha<<<(NEDGE + 7) / 8, 256, 0, stream>>>(gsrc, gdst, qb, kb, eebuf, alpha,
                                                    amax, NEDGE);
    edge_accum<<<(NEDGE + 7) / 8, 256, 0, stream>>>(gsrc, gdst, vb, eebuf, alpha, amax,
                                                    asum, hsum, NEDGE);
    combine_out<<<(NNODE * 128 + 255) / 256, 256, 0, stream>>>(hsum, asum, skip, hprev);
  }

  // ---- phase 3: final linear ----
  wmma_gemm_bias<64><<<NNODE / 16, 256, 0, stream>>>(hprev, lin_w, lin_b, (float*)d_out,
                                                     128);
}